// LSTM_23940147708543
// MI455X (gfx1250) — compile-verified
//
#include <hip/hip_runtime.h>
#include <hip/hip_bf16.h>

// ---------------------------------------------------------------------------
// 2-layer LSTM, T=512, B=64, D=H=1024, fp32, gate order i,f,g,o.
//  Phase A (per layer): xW = X @ W_ih + b   -- large parallel WMMA-f32 GEMM
//  Phase B (per layer): 512 sequential fused step kernels:
//      gates = xW[t] + h_{t-1} @ W_hh ; LSTM cell update fused in registers.
//      W_hh tiles streamed into LDS by the Tensor Data Mover (TDM),
//      double-buffered, pipelined with TENSORcnt.
// Matrix op: V_WMMA_F32_16X16X4_F32 (CDNA5 fp32 WMMA), wave32.
// ---------------------------------------------------------------------------

typedef __attribute__((ext_vector_type(2))) float v2f;
typedef __attribute__((ext_vector_type(4))) float f4;
typedef __attribute__((ext_vector_type(8))) float v8f;
typedef __attribute__((ext_vector_type(4))) unsigned int v4u;
typedef __attribute__((ext_vector_type(8))) int v8i_t;
typedef __attribute__((ext_vector_type(4))) int v4i_t;

__device__ __forceinline__ v8f wmma_f32(v2f a, v2f b, v8f c) {
  // 8 args: (neg_a, A, neg_b, B, c_mod, C, reuse_a, reuse_b)
  return __builtin_amdgcn_wmma_f32_16x16x4_f32(false, a, false, b, (short)0, c,
                                               false, false);
}

__device__ __forceinline__ float sigmoidf_fast(float x) {
  return 1.0f / (1.0f + __expf(-x));
}

// ---------------------------------------------------------------------------
// TDM: issue tensor_load_to_lds for a 3D fp32 tile.
//   X dim: tile0 contiguous elements, Y dim: tile1 lines (stride0 elements),
//   Z dim: tile2 planes (stride1 elements).  LDS filled contiguously in
//   x-fastest order starting at lds_addr.  D# packing per ISA 08 §8.
// This toolchain exposes the 6-arg builtin:
//   (uint32x4 g0, int32x8 g1, int32x4 g2, int32x4 g3, int32x8 extra, i32 cpol)
// ---------------------------------------------------------------------------
__device__ __forceinline__ void tdm_load_3d_f32(
    unsigned lds_addr, const float* gptr, unsigned tile0, unsigned tile1,
    unsigned tile2, unsigned long long stride0, unsigned long long stride1) {
  unsigned long long ga = (unsigned long long)(uintptr_t)gptr;
  v4u g0;
  g0[0] = 1u;                                      // count=1, user descriptor
  g0[1] = lds_addr;                                // bits[63:32]  lds_addr
  g0[2] = (unsigned)(ga & 0xFFFFFFFFu);            // global_addr[31:0]
  g0[3] = (unsigned)((ga >> 32) & 0x01FFFFFFu) | (2u << 30);  // addr hi, type=2
  v8i_t g1;
  g1[0] = (int)(2u << 16);                         // data_size=2 -> 4 bytes
  g1[1] = (int)((tile0 & 0xFFFFu) << 16);          // tensor_dim0 = tile0 (lo16)
  g1[2] = (int)((tile1 & 0xFFFFu) << 16);          // tdim0 hi=0 | tensor_dim1 lo
  g1[3] = (int)((tile0 & 0xFFFFu) << 16);          // tdim1 hi=0 | tile_dim0
  g1[4] = (int)((tile1 & 0xFFFFu) | ((tile2 & 0xFFFFu) << 16));  // tile1|tile2
  g1[5] = (int)(stride0 & 0xFFFFFFFFull);          // tensor_dim0_stride[31:0]
  g1[6] = (int)(((stride0 >> 32) & 0xFFFFull) |    // stride0[47:32]
                ((stride1 & 0xFFFFull) << 16));    // stride1[15:0]
  g1[7] = (int)((stride1 >> 16) & 0xFFFFFFFFull);  // stride1[47:16]
  v4i_t g2;
  g2[0] = (int)tile2;  // tensor_dim2 = tile2
  g2[1] = 0;           // tensor_dim3
  g2[2] = 0;           // tensor_dim2_stride lo (tile_dim3 unused)
  g2[3] = 0;
  v4i_t g3;
  g3[0] = 0; g3[1] = 0; g3[2] = 0; g3[3] = 0;
  v8i_t gx;            // extra operand on 6-arg toolchain: zero-filled
  gx[0] = 0; gx[1] = 0; gx[2] = 0; gx[3] = 0;
  gx[4] = 0; gx[5] = 0; gx[6] = 0; gx[7] = 0;
  __builtin_amdgcn_tensor_load_to_lds(g0, g1, g2, g3, gx, 0);
}

// ---------------------------------------------------------------------------
// Phase A: C[M,N] = A[M,K] @ W[K,N] + bias[N]
// Workgroup: 256 threads (8 waves), tile 64(M) x 128(N); wave tile 32x32.
// Kc = 16; LDS strides padded: As stride 20 (bank-conflict-free A-frag reads),
// Bs stride 132 (16B-aligned rows for b128 LDS stores).
// ---------------------------------------------------------------------------
#define GA_KC 16
#define GA_SA 20
#define GA_SB 132

__global__ __launch_bounds__(256) void gemm_bias_kernel(
    const float* __restrict__ A, const float* __restrict__ W,
    const float* __restrict__ bias, float* __restrict__ C, int K, int N) {
  __shared__ float As[64 * GA_SA];
  __shared__ float Bs[GA_KC * GA_SB];

  const int tid = threadIdx.x;
  const int wave = tid >> 5;
  const int lane = tid & 31;
  const int mBlock = blockIdx.y * 64;
  const int nBlock = blockIdx.x * 128;
  const int mWave = (wave >> 2) * 32;  // 0 or 32
  const int nWave = (wave & 3) * 32;   // 0,32,64,96

  v8f acc[2][2] = {};

  for (int k0 = 0; k0 < K; k0 += GA_KC) {
    __syncthreads();
    {  // A tile: 64 x 16 = 256 float4, 1 per thread
      int r = tid >> 2, c4 = (tid & 3) << 2;
      f4 v = *(const f4*)(A + (size_t)(mBlock + r) * K + k0 + c4);
      *(f4*)&As[r * GA_SA + c4] = v;
    }
#pragma unroll
    for (int i = 0; i < 2; ++i) {  // B tile: 16 x 128 = 512 float4, 2/thread
      int f = tid + i * 256;
      int r = f >> 5, c4 = (f & 31) << 2;
      f4 v = *(const f4*)(W + (size_t)(k0 + r) * N + nBlock + c4);
      *(f4*)&Bs[r * GA_SB + c4] = v;
    }
    __syncthreads();

    const int row = lane & 15;
    const int kp = (lane >> 4) << 1;  // lanes 0-15: K 0,1 ; lanes 16-31: K 2,3
    const int col = lane & 15;
#pragma unroll
    for (int kk = 0; kk < GA_KC; kk += 4) {
      v2f a0 = *(const v2f*)&As[(mWave + row) * GA_SA + kk + kp];
      v2f a1 = *(const v2f*)&As[(mWave + 16 + row) * GA_SA + kk + kp];
      v2f b0, b1;
      b0.x = Bs[(kk + kp) * GA_SB + nWave + col];
      b0.y = Bs[(kk + kp + 1) * GA_SB + nWave + col];
      b1.x = Bs[(kk + kp) * GA_SB + nWave + 16 + col];
      b1.y = Bs[(kk + kp + 1) * GA_SB + nWave + 16 + col];
      acc[0][0] = wmma_f32(a0, b0, acc[0][0]);
      acc[0][1] = wmma_f32(a0, b1, acc[0][1]);
      acc[1][0] = wmma_f32(a1, b0, acc[1][0]);
      acc[1][1] = wmma_f32(a1, b1, acc[1][1]);
    }
  }

  // Epilogue: D layout — VGPR r: lanes 0-15 -> M=r, lanes 16-31 -> M=r+8.
  const int colL = lane & 15;
  const int rOff = (lane >> 4) * 8;
#pragma unroll
  for (int mi = 0; mi < 2; ++mi)
#pragma unroll
    for (int ni = 0; ni < 2; ++ni) {
      int col = nBlock + nWave + ni * 16 + colL;
      float bv = bias[col];
#pragma unroll
      for (int r = 0; r < 8; ++r) {
        int row = mBlock + mWave + mi * 16 + rOff + r;
        C[(size_t)row * N + col] = acc[mi][ni][r] + bv;
      }
    }
}

// ---------------------------------------------------------------------------
// Phase B: one time step.  gates[64,4096] = xWt + h_prev[64,1024] @ W_hh.
// Workgroup: 256 threads (8 waves), covers all 64 batch rows x 32 h-columns.
// Wave (w): batch tile 16*(w&3), h-col subtile 16*(w>>2).
// Each wave holds FOUR accumulators (gates i,f,g,o) at the same 16 h-cols,
// sharing one A fragment -> LSTM cell update fuses entirely in registers.
// W_hh tiles (32 k-rows x 4 gates x 32 cols) are streamed into LDS by the
// TDM, double-buffered; one TDM kept in flight (s_wait_tensorcnt 1).
// LDS B layout (TDM x-fastest fill): [k][gate][col], stride 128 floats.
// Grid = 1024/32 = 32 workgroups.
// ---------------------------------------------------------------------------
#define ST_KC 32
#define ST_SA 36
#define ST_NCHUNK (1024 / ST_KC)

__global__ __launch_bounds__(256) void lstm_step_kernel(
    const float* __restrict__ xWt,    // [64, 4096] slice at time t
    const float* __restrict__ W_hh,   // [1024, 4096]
    const float* __restrict__ h_prev, // [64, 1024]
    float* __restrict__ c,            // [64, 1024], updated in place
    float* __restrict__ h_out) {      // [64, 1024]
  __shared__ float As[64 * ST_SA];            // h tile 64 x 32 (padded)
  __shared__ float Bs[2][ST_KC * 4 * 32];     // double-buffered TDM dest

  const int H = 1024, N4 = 4096;
  const int tid = threadIdx.x;
  const int wave = tid >> 5;
  const int lane = tid & 31;
  const int nBlock = blockIdx.x * 32;
  const int mWave = (wave & 3) * 16;
  const int nWave = (wave >> 2) * 16;
  const bool leader = (wave == 0);

  v8f acc[4] = {};

  // Prologue: kick off TDM for chunk 0 -> Bs[0].
  if (leader) {
    tdm_load_3d_f32((unsigned)(uintptr_t)&Bs[0][0], W_hh + nBlock,
                    /*tile0=*/32, /*tile1=*/4, /*tile2=*/ST_KC,
                    /*stride0=*/(unsigned long long)H,
                    /*stride1=*/(unsigned long long)N4);
  }

  for (int i = 0; i < ST_NCHUNK; ++i) {
    const int k0 = i * ST_KC;
    __syncthreads();  // previous chunk's readers done before overwrite
#pragma unroll
    for (int j = 0; j < 2; ++j) {  // A: 64x32 = 512 float4, 2/thread
      int f = tid + j * 256;
      int r = f >> 3, c4 = (f & 7) << 2;
      f4 v = *(const f4*)(h_prev + (size_t)r * H + k0 + c4);
      *(f4*)&As[r * ST_SA + c4] = v;
    }
    if (leader) {
      if (i + 1 < ST_NCHUNK) {  // prefetch next chunk via TDM, keep 1 in flight
        tdm_load_3d_f32((unsigned)(uintptr_t)&Bs[(i + 1) & 1][0],
                        W_hh + (size_t)(k0 + ST_KC) * N4 + nBlock,
                        32, 4, ST_KC, (unsigned long long)H,
                        (unsigned long long)N4);
        __builtin_amdgcn_s_wait_tensorcnt(1);  // chunk i landed in LDS
      } else {
        __builtin_amdgcn_s_wait_tensorcnt(0);
      }
    }
    __syncthreads();

    const float* Bcur = &Bs[i & 1][0];
    const int row = lane & 15;
    const int kp = (lane >> 4) << 1;  // lanes 0-15: K 0,1 ; 16-31: K 2,3
    const int col = lane & 15;
#pragma unroll
    for (int kk = 0; kk < ST_KC; kk += 4) {
      v2f a = *(const v2f*)&As[(mWave + row) * ST_SA + kk + kp];
#pragma unroll
      for (int g = 0; g < 4; ++g) {
        v2f b;
        b.x = Bcur[(kk + kp) * 128 + g * 32 + nWave + col];
        b.y = Bcur[(kk + kp + 1) * 128 + g * 32 + nWave + col];
        acc[g] = wmma_f32(a, b, acc[g]);
      }
    }
  }

  // Fused LSTM cell epilogue, entirely in registers per lane.
  const int colL = lane & 15;
  const int rOff = (lane >> 4) * 8;
  const int col = nBlock + nWave + colL;  // hidden index
#pragma unroll
  for (int r = 0; r < 8; ++r) {
    int row = mWave + rOff + r;  // batch index
    float gi = acc[0][r] + xWt[(size_t)row * N4 + 0 * H + col];
    float gf = acc[1][r] + xWt[(size_t)row * N4 + 1 * H + col];
    float gg = acc[2][r] + xWt[(size_t)row * N4 + 2 * H + col];
    float go = acc[3][r] + xWt[(size_t)row * N4 + 3 * H + col];
    float iv = sigmoidf_fast(gi);
    float fv = sigmoidf_fast(gf);
    float gv = tanhf(gg);
    float ov = sigmoidf_fast(go);
    float cold = c[(size_t)row * H + col];
    float cnew = fv * cold + iv * gv;
    float hnew = ov * tanhf(cnew);
    c[(size_t)row * H + col] = cnew;
    h_out[(size_t)row * H + col] = hnew;
  }
}

__global__ void zero_kernel(float* __restrict__ p, int n) {
  int i = blockIdx.x * 256 + threadIdx.x;
  if (i < n) p[i] = 0.0f;
}

extern "C" void kernel_launch(void* const* d_in, const int* in_sizes, int n_in,
                              void* d_out, int out_size, void* d_ws,
                              size_t ws_size, hipStream_t stream) {
  const float* x     = (const float*)d_in[0];
  const float* W_ih0 = (const float*)d_in[1];
  const float* W_hh0 = (const float*)d_in[2];
  const float* b0    = (const float*)d_in[3];
  const float* W_ih1 = (const float*)d_in[4];
  const float* W_hh1 = (const float*)d_in[5];
  const float* b1    = (const float*)d_in[6];
  float* out = (float*)d_out;

  const int T = 512, B = 64, D = 1024, H = 1024;
  const size_t BH = (size_t)B * H;

  // Workspace layout (floats): needs ~641 MiB (~672 MB).
  float* ws   = (float*)d_ws;
  float* xW   = ws;                         // T*B*4H (reused by both layers)
  float* out0 = xW + (size_t)T * B * 4 * H; // T*B*H (layer-0 hidden sequence)
  float* zb   = out0 + (size_t)T * BH;      // B*H zeros (h_{-1})
  float* c0   = zb + BH;                    // B*H
  float* c1   = c0 + BH;                    // B*H

  // zero h_{-1}, c0, c1 (contiguous 3*BH block)
  {
    int n = (int)(3 * BH);
    zero_kernel<<<(n + 255) / 256, 256, 0, stream>>>(zb, n);
  }

  dim3 gridG(4 * H / 128, (T * B) / 64);

  // -------- layer 0 --------
  gemm_bias_kernel<<<gridG, 256, 0, stream>>>(x, W_ih0, b0, xW, D, 4 * H);
  for (int t = 0; t < T; ++t) {
    const float* hp = t ? (out0 + (size_t)(t - 1) * BH) : zb;
    lstm_step_kernel<<<dim3(H / 32), 256, 0, stream>>>(
        xW + (size_t)t * B * 4 * H, W_hh0, hp, c0, out0 + (size_t)t * BH);
  }

  // -------- layer 1 (sequence output goes straight into d_out) --------
  gemm_bias_kernel<<<gridG, 256, 0, stream>>>(out0, W_ih1, b1, xW, H, 4 * H);
  for (int t = 0; t < T; ++t) {
    const float* hp = t ? (out + (size_t)(t - 1) * BH) : zb;
    lstm_step_kernel<<<dim3(H / 32), 256, 0, stream>>>(
        xW + (size_t)t * B * 4 * H, W_hh1, hp, c1, out + (size_t)t * BH);
  }

  // -------- final states: h_n [2,B,H] then c_n [2,B,H] after out1 --------
  float* hn = out + (size_t)T * BH;
  (void)hipMemcpyAsync(hn, out0 + (size_t)(T - 1) * BH, BH * sizeof(float),
                       hipMemcpyDeviceToDevice, stream);
  (void)hipMemcpyAsync(hn + BH, out + (size_t)(T - 1) * BH, BH * sizeof(float),
                       hipMemcpyDeviceToDevice, stream);
  (void)hipMemcpyAsync(hn + 2 * BH, c0, BH * sizeof(float),
                       hipMemcpyDeviceToDevice, stream);
  (void)hipMemcpyAsync(hn + 3 * BH, c1, BH * sizeof(float),
                       hipMemcpyDeviceToDevice, stream);
}